// GraphSAGERegressor_76149770158552
// MI455X (gfx1250) — compile-verified
//
#include <hip/hip_runtime.h>

typedef __attribute__((ext_vector_type(2))) float v2f;
typedef __attribute__((ext_vector_type(8))) float v8f;

#define D_IN   128
#define D_OUT  64

// ---------------------------------------------------------------------------
// Kernel 1: h0 = relu(x)  (float4), zero agg and cnt.
// ---------------------------------------------------------------------------
__global__ void k_init(const float* __restrict__ x,
                       float* __restrict__ h0,
                       float* __restrict__ agg,
                       float* __restrict__ cnt,
                       int n4 /* N*128/4 */, int c4 /* N/4 */) {
    int t = blockIdx.x * blockDim.x + threadIdx.x;
    if (t < n4) {
        float4 v = ((const float4*)x)[t];
        float4 r;
        r.x = fmaxf(v.x, 0.0f);
        r.y = fmaxf(v.y, 0.0f);
        r.z = fmaxf(v.z, 0.0f);
        r.w = fmaxf(v.w, 0.0f);
        ((float4*)h0)[t] = r;
        ((float4*)agg)[t] = make_float4(0.0f, 0.0f, 0.0f, 0.0f);
    }
    if (t < c4) {
        ((float4*)cnt)[t] = make_float4(0.0f, 0.0f, 0.0f, 0.0f);
    }
}

// ---------------------------------------------------------------------------
// Kernel 2: edge scatter. One wave32 per edge; lane L handles features
// [4L, 4L+3]: float4 gather from h0[src], 4 native f32 atomic adds into
// agg[dst]. Lane 0 bumps the (float) in-degree count.
// ---------------------------------------------------------------------------
__global__ __launch_bounds__(256)
void k_scatter(const int* __restrict__ src,
               const int* __restrict__ dst,
               const float* __restrict__ h0,
               float* __restrict__ agg,
               float* __restrict__ cnt,
               int nedges) {
    const int lane = threadIdx.x & 31;
    const int e = blockIdx.x * (blockDim.x >> 5) + (threadIdx.x >> 5);
    if (e >= nedges) return;

    const int s = src[e];
    const int d = dst[e];

    float4 v = ((const float4*)(h0 + (size_t)s * D_IN))[lane];
    float* out = agg + (size_t)d * D_IN + 4 * lane;
    __hip_atomic_fetch_add(out + 0, v.x, __ATOMIC_RELAXED, __HIP_MEMORY_SCOPE_AGENT);
    __hip_atomic_fetch_add(out + 1, v.y, __ATOMIC_RELAXED, __HIP_MEMORY_SCOPE_AGENT);
    __hip_atomic_fetch_add(out + 2, v.z, __ATOMIC_RELAXED, __HIP_MEMORY_SCOPE_AGENT);
    __hip_atomic_fetch_add(out + 3, v.w, __ATOMIC_RELAXED, __HIP_MEMORY_SCOPE_AGENT);
    if (lane == 0) {
        __hip_atomic_fetch_add(cnt + d, 1.0f, __ATOMIC_RELAXED, __HIP_MEMORY_SCOPE_AGENT);
    }
}

// ---------------------------------------------------------------------------
// Kernel 3: fused SAGE GEMM with V_WMMA_F32_16X16X4_F32.
//   hout[i, n] = relu( mean[i,:] @ Wl[n,:] + bl[n] + h0[i,:] @ Wr[n,:] )
// Block = 128 threads (4 waves). Block owns 16 nodes; wave w owns output
// columns [16w, 16w+15]. K=128 in steps of 4 (32 WMMAs per GEMM, 64 total).
//
// f32 WMMA lane layout (ISA 7.12.2):
//   A 16x4 : lane = m + 16*kh holds {K=2kh, K=2kh+1} of row m
//   B 4x16 : lane = n + 16*kh holds {K=2kh, K=2kh+1} of col n
//   C 16x16: VGPR v at lane holds row (v + 8*kh), col (lane&15)
// ---------------------------------------------------------------------------
__global__ __launch_bounds__(128)
void k_sage_gemm(const float* __restrict__ agg,
                 const float* __restrict__ cnt,
                 const float* __restrict__ h0,
                 const float* __restrict__ Wl,   // [64,128]
                 const float* __restrict__ bl,   // [64]
                 const float* __restrict__ Wr,   // [64,128]
                 float* __restrict__ hout) {     // [N,64]
    const int lane = threadIdx.x & 31;
    const int wave = threadIdx.x >> 5;           // 0..3
    const int m    = lane & 15;                  // A row / B,C column (local)
    const int kh   = lane >> 4;                  // K-half selector
    const int node = blockIdx.x * 16 + m;        // A row (global node id)
    const int n    = wave * 16 + m;              // output feature column

    const float c   = cnt[node];
    const float scl = 1.0f / fmaxf(c, 1.0f);     // mean = agg * scl

    const float* __restrict__ arow = agg + (size_t)node * D_IN;
    const float* __restrict__ hrow = h0  + (size_t)node * D_IN;
    const float* __restrict__ wlr  = Wl  + (size_t)n * D_IN;
    const float* __restrict__ wrr  = Wr  + (size_t)n * D_IN;

    v8f acc = {};

    // GEMM 1: mean @ Wl.T
#pragma unroll
    for (int kb = 0; kb < D_IN; kb += 4) {
        const int k0 = kb + 2 * kh;
        v2f a; a.x = arow[k0] * scl; a.y = arow[k0 + 1] * scl;
        v2f b; b.x = wlr[k0];        b.y = wlr[k0 + 1];
        acc = __builtin_amdgcn_wmma_f32_16x16x4_f32(
            false, a, false, b, (short)0, acc, false, false);
    }
    // GEMM 2: h0 @ Wr.T (accumulated into same C tile)
#pragma unroll
    for (int kb = 0; kb < D_IN; kb += 4) {
        const int k0 = kb + 2 * kh;
        v2f a; a.x = hrow[k0]; a.y = hrow[k0 + 1];
        v2f b; b.x = wrr[k0];  b.y = wrr[k0 + 1];
        acc = __builtin_amdgcn_wmma_f32_16x16x4_f32(
            false, a, false, b, (short)0, acc, false, false);
    }

    const float bias = bl[n];
#pragma unroll
    for (int v = 0; v < 8; ++v) {
        const float r = fmaxf(acc[v] + bias, 0.0f);
        const int row = blockIdx.x * 16 + v + 8 * kh;
        hout[(size_t)row * D_OUT + n] = r;
    }
}

// ---------------------------------------------------------------------------
// Kernel 4: y[i] = h[i,:] . W_head + b_head
// ---------------------------------------------------------------------------
__global__ void k_head(const float* __restrict__ h,     // [N,64]
                       const float* __restrict__ Wh,    // [64]
                       const float* __restrict__ bh,    // [1]
                       float* __restrict__ y,           // [N]
                       int nnodes) {
    int i = blockIdx.x * blockDim.x + threadIdx.x;
    if (i >= nnodes) return;
    const float4* hr = (const float4*)(h + (size_t)i * D_OUT);
    const float4* wr = (const float4*)Wh;
    float acc = 0.0f;
#pragma unroll
    for (int j = 0; j < D_OUT / 4; ++j) {
        float4 a = hr[j];
        float4 b = wr[j];
        acc += a.x * b.x + a.y * b.y + a.z * b.z + a.w * b.w;
    }
    y[i] = acc + bh[0];
}

// ---------------------------------------------------------------------------
// Launch
// ---------------------------------------------------------------------------
extern "C" void kernel_launch(void* const* d_in, const int* in_sizes, int n_in,
                              void* d_out, int out_size, void* d_ws, size_t ws_size,
                              hipStream_t stream) {
    const float* x       = (const float*)d_in[0];   // [N,128]
    const int*   ei      = (const int*)d_in[1];     // [2,E]
    const float* W2_l    = (const float*)d_in[2];   // [64,128]
    const float* b2_l    = (const float*)d_in[3];   // [64]
    const float* W2_r    = (const float*)d_in[4];   // [64,128]
    const float* W_head  = (const float*)d_in[5];   // [64]
    const float* b_head  = (const float*)d_in[6];   // [1]

    const int N = in_sizes[0] / D_IN;               // 100000
    const int E = in_sizes[1] / 2;                  // 1600000

    const int* src = ei;
    const int* dst = ei + E;

    // Workspace layout
    char* ws   = (char*)d_ws;
    float* h0  = (float*)(ws);                                      // N*128 f32
    float* agg = (float*)(ws + (size_t)N * D_IN * sizeof(float));   // N*128 f32
    float* cnt = (float*)(ws + (size_t)2 * N * D_IN * sizeof(float)); // N f32

    // Output layout: [h (N*64), y (N)]
    float* hout = (float*)d_out;
    float* y    = hout + (size_t)N * D_OUT;

    // 1) relu + zero
    {
        const int n4 = N * D_IN / 4;   // 3,200,000
        const int c4 = (N + 3) / 4;    // 25,000
        const int grid = (n4 + 255) / 256;
        k_init<<<grid, 256, 0, stream>>>(x, h0, agg, cnt, n4, c4);
    }
    // 2) edge scatter (8 edges per 256-thread block)
    {
        const int grid = (E + 7) / 8;  // 200,000
        k_scatter<<<grid, 256, 0, stream>>>(src, dst, h0, agg, cnt, E);
    }
    // 3) fused SAGE GEMM (WMMA), 16 nodes per block
    {
        const int grid = N / 16;       // 6,250 (N divisible by 16)
        k_sage_gemm<<<grid, 128, 0, stream>>>(agg, cnt, h0, W2_l, b2_l, W2_r, hout);
    }
    // 4) head
    {
        const int grid = (N + 255) / 256;
        k_head<<<grid, 256, 0, stream>>>(hout, W_head, b_head, y, N);
    }
}